// GGNN_8340826489019
// MI455X (gfx1250) — compile-verified
//
#include <hip/hip_runtime.h>
#include <math.h>

#define D 128
#define LDSP 132   // padded LDS row stride (floats) to avoid bank conflicts

typedef __attribute__((ext_vector_type(2))) float v2f;
typedef __attribute__((ext_vector_type(8))) float v8f;

#define WMMA_F32(a, b, c) \
  __builtin_amdgcn_wmma_f32_16x16x4_f32(false, (a), false, (b), (short)0, (c), false, false)

__device__ __forceinline__ float sigm(float x) { return 1.0f / (1.0f + __expf(-x)); }

// ---------------------------------------------------------------------------
// h[n][d] = embed_table[x[n]][d]
__global__ __launch_bounds__(256) void embed_kernel(const int* __restrict__ x,
                                                    const float* __restrict__ tab,
                                                    float* __restrict__ h, int N) {
  int i = blockIdx.x * 256 + threadIdx.x;           // one float4 per thread
  int total = N * (D / 4);
  if (i >= total) return;
  int n  = i >> 5;                                   // 32 float4 per row
  int k4 = (i & 31) * 4;
  *(float4*)(h + (size_t)n * D + k4) =
      *(const float4*)(tab + (size_t)x[n] * D + k4);
}

// ---------------------------------------------------------------------------
__global__ void zero_kernel(float* __restrict__ p, long long count) {
  long long i = (long long)blockIdx.x * blockDim.x + threadIdx.x;
  long long stride = (long long)gridDim.x * blockDim.x;
  for (; i < count; i += stride) p[i] = 0.0f;
}

// ---------------------------------------------------------------------------
// Pre-pack layer weights W[l] (D x D, B[k][n] = W[k*D+n]) into WMMA-fragment
// order: Wp[l][((ct*32 + s)*32 + lane)*2 + {0,1}] = {B[k0][col], B[k0+1][col]}
// with k0 = 4s + 2*(lane>>4), col = 16*ct + (lane&15).
// Reader then does ONE coalesced b64 load per wmma step.
__global__ __launch_bounds__(256) void pack_gemmW(const float* __restrict__ W,
                                                  float* __restrict__ Wp, int L) {
  int idx = blockIdx.x * 256 + threadIdx.x;
  int total = L * 8192;                     // 8 ct * 32 s * 32 lanes
  if (idx >= total) return;
  int l    = idx / 8192;
  int r    = idx % 8192;
  int ct   = r >> 10;
  int s    = (r >> 5) & 31;
  int lane = r & 31;
  int mrow = lane & 15, khalf = lane >> 4;
  int k0   = 4 * s + 2 * khalf;
  int col  = ct * 16 + mrow;
  const float* Wl = W + (size_t)l * D * D;
  float2 v;
  v.x = Wl[(size_t)k0 * D + col];
  v.y = Wl[(size_t)(k0 + 1) * D + col];
  *(float2*)(Wp + (size_t)l * D * D + (size_t)r * 2) = v;
}

// Pre-pack GRU weights. mat 0..2 = W_ih r/z/n, mat 3..5 = W_hh r/z/n.
// B[k][n] = Wsrc[jcol*D + k] with jcol = (mat%3)*D + 16*ct + (lane&15).
__global__ __launch_bounds__(256) void pack_gruW(const float* __restrict__ Wih,
                                                 const float* __restrict__ Whh,
                                                 float* __restrict__ Gp) {
  int idx = blockIdx.x * 256 + threadIdx.x;
  if (idx >= 6 * 8192) return;
  int mat  = idx / 8192;
  int r    = idx % 8192;
  int ct   = r >> 10;
  int s    = (r >> 5) & 31;
  int lane = r & 31;
  int mrow = lane & 15, khalf = lane >> 4;
  int k0   = 4 * s + 2 * khalf;
  int jcol = (mat % 3) * D + ct * 16 + mrow;
  const float* Wsrc = (mat < 3) ? Wih : Whh;
  float2 v = *(const float2*)(Wsrc + (size_t)jcol * D + k0);
  *(float2*)(Gp + (size_t)mat * D * D + (size_t)r * 2) = v;
}

// ---------------------------------------------------------------------------
// m = h @ Wl  with fragment-packed weights.
// Block = 256 thr = 8 waves; block owns 32 rows (two 16-row tiles), wave w owns
// cols [16w,16w+16). Each B fragment (one b64 load) feeds TWO wmma ops.
__global__ __launch_bounds__(256) void gemm_hW(const float* __restrict__ h,
                                               const float* __restrict__ Wp,
                                               float* __restrict__ m, int N) {
  __shared__ float sA[32 * LDSP];
  const int tid = threadIdx.x;
  const int rowBase = blockIdx.x * 32;
  for (int i = tid; i < 1024; i += 256) {   // 32 rows x 32 float4
    int r  = i >> 5;
    int k4 = (i & 31) * 4;
    int row = rowBase + r; if (row >= N) row = N - 1;
    *(float4*)(sA + r * LDSP + k4) = *(const float4*)(h + (size_t)row * D + k4);
  }
  __syncthreads();

  const int lane  = tid & 31;
  const int wave  = tid >> 5;
  const int mrow  = lane & 15;
  const int khalf = lane >> 4;
  const int col   = wave * 16 + mrow;
  const float* wp = Wp + (size_t)wave * 2048 + lane * 2;  // + s*64 per step

  v8f acc0 = {}, acc1 = {};
  for (int s = 0; s < 32; ++s) {
    int k0 = 4 * s + 2 * khalf;
    v2f b = *(const v2f*)(wp + s * 64);
    v2f a0; a0.x = sA[mrow * LDSP + k0];        a0.y = sA[mrow * LDSP + k0 + 1];
    v2f a1; a1.x = sA[(16 + mrow) * LDSP + k0]; a1.y = sA[(16 + mrow) * LDSP + k0 + 1];
    acc0 = WMMA_F32(a0, b, acc0);
    acc1 = WMMA_F32(a1, b, acc1);
  }
  const int rOff = khalf * 8;
#pragma unroll
  for (int j = 0; j < 8; ++j) {
    int row0 = rowBase + j + rOff;
    int row1 = rowBase + 16 + j + rOff;
    if (row0 < N) m[(size_t)row0 * D + col] = acc0[j];
    if (row1 < N) m[(size_t)row1 * D + col] = acc1[j];
  }
}

// ---------------------------------------------------------------------------
// For each edge e: agg[dst[e]] += m[src[e]] * ew[e]   (one wave32 per edge)
__global__ __launch_bounds__(256) void edge_kernel(const int* __restrict__ src,
                                                   const int* __restrict__ dst,
                                                   const int* __restrict__ eattr,
                                                   const float* __restrict__ m,
                                                   float* __restrict__ agg, int E) {
  int e    = blockIdx.x * 8 + (threadIdx.x >> 5);
  int lane = threadIdx.x & 31;
  if (e >= E) return;
  int s = src[e], d = dst[e];
  float w = (float)eattr[e];
  float4 v = *(const float4*)(m + (size_t)s * D + lane * 4);
  float* ap = agg + (size_t)d * D + lane * 4;
  unsafeAtomicAdd(ap + 0, v.x * w);
  unsafeAtomicAdd(ap + 1, v.y * w);
  unsafeAtomicAdd(ap + 2, v.z * w);
  unsafeAtomicAdd(ap + 3, v.w * w);
}

// ---------------------------------------------------------------------------
// GRU cell with fragment-packed weights: gi = agg @ W_ih^T + b_ih ;
// gh = h @ W_hh^T + b_hh ; gate math ; h updated in place.
// Wave holds 6 accumulators (48 VGPRs); 192 WMMAs; 6 coalesced b64 B-loads/step.
__global__ __launch_bounds__(256) void gru_kernel(const float* __restrict__ agg,
                                                  float* __restrict__ h,
                                                  const float* __restrict__ Gp,
                                                  const float* __restrict__ bih,
                                                  const float* __restrict__ bhh, int N) {
  __shared__ float sAgg[16 * LDSP];
  __shared__ float sH[16 * LDSP];
  const int tid = threadIdx.x;
  const int rowBase = blockIdx.x * 16;
  for (int i = tid; i < 512; i += 256) {
    int r  = i >> 5;
    int k4 = (i & 31) * 4;
    int row = rowBase + r; if (row >= N) row = N - 1;
    *(float4*)(sAgg + r * LDSP + k4) = *(const float4*)(agg + (size_t)row * D + k4);
    *(float4*)(sH   + r * LDSP + k4) = *(const float4*)(h   + (size_t)row * D + k4);
  }
  __syncthreads();

  const int lane  = tid & 31;
  const int wave  = tid >> 5;
  const int mrow  = lane & 15;
  const int khalf = lane >> 4;
  const int col   = wave * 16 + mrow;
  const float* gp = Gp + (size_t)wave * 2048 + lane * 2;  // + mat*16384 + s*64

  v8f cir = {}, ciz = {}, cin = {}, chr_ = {}, chz = {}, chn = {};
  for (int s = 0; s < 32; ++s) {
    int k0 = 4 * s + 2 * khalf;
    v2f aA; aA.x = sAgg[mrow * LDSP + k0]; aA.y = sAgg[mrow * LDSP + k0 + 1];
    v2f aH; aH.x = sH[mrow * LDSP + k0];   aH.y = sH[mrow * LDSP + k0 + 1];
    const float* gs = gp + s * 64;
    v2f bIR = *(const v2f*)(gs);
    v2f bIZ = *(const v2f*)(gs + 1 * 16384);
    v2f bIN = *(const v2f*)(gs + 2 * 16384);
    v2f bHR = *(const v2f*)(gs + 3 * 16384);
    v2f bHZ = *(const v2f*)(gs + 4 * 16384);
    v2f bHN = *(const v2f*)(gs + 5 * 16384);
    cir  = WMMA_F32(aA, bIR, cir);
    ciz  = WMMA_F32(aA, bIZ, ciz);
    cin  = WMMA_F32(aA, bIN, cin);
    chr_ = WMMA_F32(aH, bHR, chr_);
    chz  = WMMA_F32(aH, bHZ, chz);
    chn  = WMMA_F32(aH, bHN, chn);
  }

  const float bir = bih[col], biz = bih[D + col], bin = bih[2 * D + col];
  const float bhr = bhh[col], bhz = bhh[D + col], bhn = bhh[2 * D + col];
  const int rOff = khalf * 8;
#pragma unroll
  for (int j = 0; j < 8; ++j) {
    int row = rowBase + j + rOff;
    if (row < N) {
      float r  = sigm(cir[j] + bir + chr_[j] + bhr);
      float z  = sigm(ciz[j] + biz + chz[j] + bhz);
      float nn = tanhf(cin[j] + bin + r * (chn[j] + bhn));
      float hv = sH[(j + rOff) * LDSP + col];
      h[(size_t)row * D + col] = (1.0f - z) * nn + z * hv;
    }
  }
}

// ---------------------------------------------------------------------------
// e[n] = exp(sigmoid(h[n].gate_w + gate_b));  sumExp += block partial
// (sigmoid in (0,1) => exp needs no max-subtraction: numerically safe)
__global__ __launch_bounds__(256) void gate_kernel(const float* __restrict__ h,
                                                   const float* __restrict__ gw,
                                                   const float* __restrict__ gb,
                                                   float* __restrict__ e,
                                                   float* __restrict__ sumExp, int N) {
  __shared__ float red[256];
  int n = blockIdx.x * 256 + threadIdx.x;
  float ev = 0.0f;
  if (n < N) {
    float acc = 0.0f;
    const float4* hp = (const float4*)(h + (size_t)n * D);
    const float4* gp = (const float4*)gw;
#pragma unroll 8
    for (int i = 0; i < 32; ++i) {
      float4 a = hp[i], b = gp[i];
      acc += a.x * b.x + a.y * b.y + a.z * b.z + a.w * b.w;
    }
    ev = __expf(sigm(acc + gb[0]));
    e[n] = ev;
  }
  red[threadIdx.x] = ev;
  __syncthreads();
  for (int s = 128; s > 0; s >>= 1) {
    if (threadIdx.x < s) red[threadIdx.x] += red[threadIdx.x + s];
    __syncthreads();
  }
  if (threadIdx.x == 0) unsafeAtomicAdd(sumExp, red[0]);
}

// hg[d] += sum over node chunk of e[n]*h[n][d]
__global__ __launch_bounds__(128) void pool_kernel(const float* __restrict__ h,
                                                   const float* __restrict__ e,
                                                   float* __restrict__ hg, int N) {
  int d  = threadIdx.x;
  int n0 = blockIdx.x * 256;
  int n1 = n0 + 256; if (n1 > N) n1 = N;
  float acc = 0.0f;
  for (int n = n0; n < n1; ++n) acc += e[n] * h[(size_t)n * D + d];
  unsafeAtomicAdd(hg + d, acc);
}

// out[j] = (hg . out_w[j]) / sumExp + out_b[j]
__global__ void out_kernel(const float* __restrict__ hg, const float* __restrict__ sumExp,
                           const float* __restrict__ ow, const float* __restrict__ ob,
                           float* __restrict__ out) {
  if (threadIdx.x == 0 && blockIdx.x == 0) {
    float inv = 1.0f / sumExp[0];
    for (int j = 0; j < 2; ++j) {
      float acc = 0.0f;
      for (int d = 0; d < D; ++d) acc += hg[d] * ow[j * D + d];
      out[j] = inv * acc + ob[j];
    }
  }
}

// ---------------------------------------------------------------------------
extern "C" void kernel_launch(void* const* d_in, const int* in_sizes, int n_in,
                              void* d_out, int out_size, void* d_ws, size_t ws_size,
                              hipStream_t stream) {
  const int*   x    = (const int*)d_in[0];
  const int*   ei   = (const int*)d_in[1];
  const int*   ea   = (const int*)d_in[2];
  const float* tab  = (const float*)d_in[3];
  const float* W    = (const float*)d_in[4];
  const float* Wih  = (const float*)d_in[5];
  const float* Whh  = (const float*)d_in[6];
  const float* bih  = (const float*)d_in[7];
  const float* bhh  = (const float*)d_in[8];
  const float* gw   = (const float*)d_in[9];
  const float* gb   = (const float*)d_in[10];
  const float* ow   = (const float*)d_in[11];
  const float* ob   = (const float*)d_in[12];

  const int N = in_sizes[0];
  const int E = in_sizes[2];
  const int L = in_sizes[4] / (D * D);

  float* h      = (float*)d_ws;
  float* m      = h   + (size_t)N * D;
  float* agg    = m   + (size_t)N * D;
  float* e      = agg + (size_t)N * D;
  float* hg     = e   + N;
  float* sumExp = hg  + D;              // 1 float
  float* Wp     = sumExp + 1;           // L*D*D packed layer weights
  float* Gp     = Wp + (size_t)L * D * D;  // 6*D*D packed GRU weights

  const int* src = ei;
  const int* dst = ei + E;

  const int nblk32  = (N + 31) / 32;
  const int nblk16  = (N + 15) / 16;
  const int embBlks = (N * (D / 4) + 255) / 256;

  pack_gemmW<<<(L * 8192 + 255) / 256, 256, 0, stream>>>(W, Wp, L);
  pack_gruW<<<(6 * 8192 + 255) / 256, 256, 0, stream>>>(Wih, Whh, Gp);
  embed_kernel<<<embBlks, 256, 0, stream>>>(x, tab, h, N);

  for (int l = 0; l < L; ++l) {
    gemm_hW<<<nblk32, 256, 0, stream>>>(h, Wp + (size_t)l * D * D, m, N);
    zero_kernel<<<2048, 256, 0, stream>>>(agg, (long long)N * D);
    edge_kernel<<<(E + 7) / 8, 256, 0, stream>>>(src, dst, ea, m, agg, E);
    gru_kernel<<<nblk16, 256, 0, stream>>>(agg, h, Gp, bih, bhh, N);
  }

  zero_kernel<<<1, 256, 0, stream>>>(hg, (long long)(D + 1));  // hg + sumExp
  gate_kernel<<<(N + 255) / 256, 256, 0, stream>>>(h, gw, gb, e, sumExp, N);
  pool_kernel<<<(N + 255) / 256, 128, 0, stream>>>(h, e, hg, N);
  out_kernel<<<1, 64, 0, stream>>>(hg, sumExp, ow, ob, (float*)d_out);
}